// GATNet_88630945120477
// MI455X (gfx1250) — compile-verified
//
#include <hip/hip_runtime.h>
#include <math.h>

typedef __attribute__((ext_vector_type(2))) float v2f;
typedef __attribute__((ext_vector_type(8))) float v8f;

#define FIN   128
#define HID   8
#define NCLS  16
#define NEG_SLOPE 0.2f

// -------- float atomic max (sign-split monotonic int trick; m initialized to -inf)
__device__ __forceinline__ void atomicMaxF(float* addr, float v) {
    if (v >= 0.0f) atomicMax((int*)addr, __float_as_int(v));
    else           atomicMin((unsigned int*)addr, __float_as_uint(v));
}

// -------- workspace init: segment-max to -inf, denoms/accumulators to zero
__global__ void init_ws(float* m1, float* dn1, float* agg1,
                        float* m2, float* dn2, float* agg2, int n) {
    int i = blockIdx.x * blockDim.x + threadIdx.x;
    if (i < n * NCLS) agg2[i] = 0.0f;
    if (i < n * HID)  agg1[i] = 0.0f;
    if (i < n) {
        m1[i] = -__builtin_inff();
        m2[i] = -__builtin_inff();
        dn1[i] = 0.0f;
        dn2[i] = 0.0f;
    }
}

// -------- layer-1 GEMM: h1[N,8] = x[N,128] @ W1[128,8], via V_WMMA_F32_16X16X4_F32
// one wave per 16-row tile; 4 waves per block; W1 staged zero-padded (128x16) in LDS
__global__ void gemm1_wmma(const float* __restrict__ x, const float* __restrict__ W1,
                           float* __restrict__ h1, int n) {
    __shared__ float wlds[FIN * 16];
    int t = threadIdx.x;                    // 128 threads
    {   // stage padded W1: row k of 16 cols (cols >= HID are zero)
        int k = t;                          // 0..127
        const float* wr = W1 + k * HID;
        #pragma unroll
        for (int c = 0; c < 16; ++c) wlds[k * 16 + c] = (c < HID) ? wr[c] : 0.0f;
    }
    __syncthreads();

    int wave = t >> 5, lane = t & 31;
    int tile = blockIdx.x * 4 + wave;
    int col  = lane & 15;
    int hi   = lane >> 4;                    // 0 or 1
    int mrow = tile * 16 + col;              // A-row this lane feeds
    if (mrow >= n) mrow = n - 1;             // clamp (n % 16 == 0 for N=100000)
    const float* xrow = x + (long)mrow * FIN;

    v8f c = {};
    #pragma unroll
    for (int k0 = 0; k0 < FIN; k0 += 4) {
        int ka = k0 + 2 * hi;
        v2f a = *(const v2f*)(xrow + ka);    // A: {K=ka, K=ka+1} for this lane's M
        v2f b;
        b.x = wlds[ka * 16 + col];           // B: row K=ka,   col N=lane&15
        b.y = wlds[(ka + 1) * 16 + col];     // B: row K=ka+1
        c = __builtin_amdgcn_wmma_f32_16x16x4_f32(false, a, false, b,
                                                  (short)0, c, false, false);
    }

    // D: VGPR j -> M = j + 8*hi, N = lane&15
    int mbase = tile * 16 + (hi << 3);
    if (col < HID) {
        #pragma unroll
        for (int j = 0; j < 8; ++j) {
            int r = mbase + j;
            if (r < n) h1[(long)r * HID + col] = c[j];
        }
    }
}

// -------- layer-2 GEMM: h2[N,16] = h1r[N,8] @ W2[8,16]
__global__ void gemm2_wmma(const float* __restrict__ h1r, const float* __restrict__ W2,
                           float* __restrict__ h2, int n) {
    __shared__ float wlds[HID * 16];         // 8x16, no padding needed
    int t = threadIdx.x;                     // 128 threads
    if (t < HID * 16) wlds[t] = W2[t];
    __syncthreads();

    int wave = t >> 5, lane = t & 31;
    int tile = blockIdx.x * 4 + wave;
    int col  = lane & 15;
    int hi   = lane >> 4;
    int mrow = tile * 16 + col;
    if (mrow >= n) mrow = n - 1;
    const float* hrow = h1r + (long)mrow * HID;

    v8f c = {};
    #pragma unroll
    for (int k0 = 0; k0 < HID; k0 += 4) {
        int ka = k0 + 2 * hi;
        v2f a = *(const v2f*)(hrow + ka);
        v2f b;
        b.x = wlds[ka * 16 + col];
        b.y = wlds[(ka + 1) * 16 + col];
        c = __builtin_amdgcn_wmma_f32_16x16x4_f32(false, a, false, b,
                                                  (short)0, c, false, false);
    }

    int mbase = tile * 16 + (hi << 3);
    #pragma unroll
    for (int j = 0; j < 8; ++j) {
        int r = mbase + j;
        if (r < n) h2[(long)r * NCLS + col] = c[j];
    }
}

// -------- per-node attention logits: as[i] = h[i,:]·a_src, ad[i] = h[i,:]·a_dst
// F compile-time -> float4 loads, fully unrolled
template <int F>
__global__ void alpha_kernel(const float* __restrict__ h,
                             const float* __restrict__ a_src,
                             const float* __restrict__ a_dst,
                             float* __restrict__ outs, float* __restrict__ outd,
                             int n) {
    int i = blockIdx.x * blockDim.x + threadIdx.x;
    if (i >= n) return;
    const float4* hr = (const float4*)(h + (long)i * F);
    float s = 0.0f, d = 0.0f;
    #pragma unroll
    for (int q = 0; q < F / 4; ++q) {
        float4 hv = hr[q];
        s += hv.x * a_src[4 * q + 0] + hv.y * a_src[4 * q + 1]
           + hv.z * a_src[4 * q + 2] + hv.w * a_src[4 * q + 3];
        d += hv.x * a_dst[4 * q + 0] + hv.y * a_dst[4 * q + 1]
           + hv.z * a_dst[4 * q + 2] + hv.w * a_dst[4 * q + 3];
    }
    outs[i] = s;
    outd[i] = d;
}

// -------- edge pass 1: segment max of leaky_relu(as[src]+ad[dst]) keyed by dst
__global__ void edge_max(const int* __restrict__ src, const int* __restrict__ dst,
                         const float* __restrict__ as, const float* __restrict__ ad,
                         float* __restrict__ m, int E) {
    int i = blockIdx.x * blockDim.x + threadIdx.x;
    if (i >= E) return;
    int s = src[i], d = dst[i];
    float e = as[s] + ad[d];
    e = (e >= 0.0f) ? e : NEG_SLOPE * e;
    atomicMaxF(&m[d], e);
}

// -------- edge pass 2: ex = exp(e - m[dst]); denom += ex; agg[dst] += ex * h[src]
// F compile-time -> b128 gathers of h[src], unrolled global_atomic_add_f32 chain
template <int F>
__global__ void edge_acc(const int* __restrict__ src, const int* __restrict__ dst,
                         const float* __restrict__ as, const float* __restrict__ ad,
                         const float* __restrict__ m, const float* __restrict__ h,
                         float* __restrict__ denom, float* __restrict__ agg, int E) {
    int i = blockIdx.x * blockDim.x + threadIdx.x;
    if (i >= E) return;
    int s = src[i], d = dst[i];
    float e = as[s] + ad[d];
    e = (e >= 0.0f) ? e : NEG_SLOPE * e;
    float ex = __expf(e - m[d]);
    atomicAdd(&denom[d], ex);
    const float4* hs = (const float4*)(h + (long)s * F);
    float* ag = agg + (long)d * F;
    #pragma unroll
    for (int q = 0; q < F / 4; ++q) {
        float4 hv = hs[q];                   // L2-resident gather, global_load_b128
        atomicAdd(&ag[4 * q + 0], ex * hv.x);
        atomicAdd(&ag[4 * q + 1], ex * hv.y);
        atomicAdd(&ag[4 * q + 2], ex * hv.z);
        atomicAdd(&ag[4 * q + 3], ex * hv.w);
    }
}

// -------- layer-1 epilogue: agg/denom + bias, ReLU (in place: agg1 becomes layer-2 input)
__global__ void finish1(float* __restrict__ agg, const float* __restrict__ denom,
                        const float* __restrict__ bias, int n) {
    int i = blockIdx.x * blockDim.x + threadIdx.x;
    if (i >= n) return;
    float d = denom[i];
    float inv = (d > 0.0f) ? 1.0f / d : 0.0f;
    float4* ap = (float4*)(agg + (long)i * HID);
    #pragma unroll
    for (int q = 0; q < HID / 4; ++q) {
        float4 v = ap[q];
        v.x = fmaxf(v.x * inv + bias[4 * q + 0], 0.0f);
        v.y = fmaxf(v.y * inv + bias[4 * q + 1], 0.0f);
        v.z = fmaxf(v.z * inv + bias[4 * q + 2], 0.0f);
        v.w = fmaxf(v.w * inv + bias[4 * q + 3], 0.0f);
        ap[q] = v;
    }
}

// -------- layer-2 epilogue: agg/denom + bias, then log_softmax over 16 classes
__global__ void finish2(const float* __restrict__ agg, const float* __restrict__ denom,
                        const float* __restrict__ bias, float* __restrict__ out, int n) {
    int i = blockIdx.x * blockDim.x + threadIdx.x;
    if (i >= n) return;
    float d = denom[i];
    float inv = (d > 0.0f) ? 1.0f / d : 0.0f;
    const float4* ap = (const float4*)(agg + (long)i * NCLS);
    float v[NCLS];
    float mx = -__builtin_inff();
    #pragma unroll
    for (int q = 0; q < NCLS / 4; ++q) {
        float4 a = ap[q];
        v[4 * q + 0] = a.x * inv + bias[4 * q + 0];
        v[4 * q + 1] = a.y * inv + bias[4 * q + 1];
        v[4 * q + 2] = a.z * inv + bias[4 * q + 2];
        v[4 * q + 3] = a.w * inv + bias[4 * q + 3];
    }
    #pragma unroll
    for (int f = 0; f < NCLS; ++f) mx = fmaxf(mx, v[f]);
    float s = 0.0f;
    #pragma unroll
    for (int f = 0; f < NCLS; ++f) s += __expf(v[f] - mx);
    float ls = __logf(s) + mx;
    float4* op = (float4*)(out + (long)i * NCLS);
    #pragma unroll
    for (int q = 0; q < NCLS / 4; ++q) {
        float4 o;
        o.x = v[4 * q + 0] - ls;
        o.y = v[4 * q + 1] - ls;
        o.z = v[4 * q + 2] - ls;
        o.w = v[4 * q + 3] - ls;
        op[q] = o;
    }
}

extern "C" void kernel_launch(void* const* d_in, const int* in_sizes, int n_in,
                              void* d_out, int out_size, void* d_ws, size_t ws_size,
                              hipStream_t stream) {
    const float* x   = (const float*)d_in[0];
    const int*   ei  = (const int*)  d_in[1];
    const float* W1  = (const float*)d_in[2];
    const float* a1s = (const float*)d_in[3];
    const float* a1d = (const float*)d_in[4];
    const float* b1  = (const float*)d_in[5];
    const float* W2  = (const float*)d_in[6];
    const float* a2s = (const float*)d_in[7];
    const float* a2d = (const float*)d_in[8];
    const float* b2  = (const float*)d_in[9];

    int n = in_sizes[0] / FIN;            // 100000
    int E = in_sizes[1] / 2;              // 6400000
    const int* src = ei;
    const int* dst = ei + E;

    // workspace carve-out (floats): ~56*N floats ≈ 22.4 MB; every slice is 16B-aligned
    float* ws   = (float*)d_ws;
    float* h1   = ws; ws += (long)n * HID;
    float* as1  = ws; ws += n;
    float* ad1  = ws; ws += n;
    float* m1   = ws; ws += n;
    float* dn1  = ws; ws += n;
    float* agg1 = ws; ws += (long)n * HID;   // becomes relu'd layer-2 input in place
    float* h2   = ws; ws += (long)n * NCLS;
    float* as2  = ws; ws += n;
    float* ad2  = ws; ws += n;
    float* m2   = ws; ws += n;
    float* dn2  = ws; ws += n;
    float* agg2 = ws; ws += (long)n * NCLS;

    int ntiles  = (n + 15) / 16;
    int gblocks = (ntiles + 3) / 4;       // 4 waves (tiles) per 128-thread block
    int eblocks = (E + 255) / 256;
    int nblocks = (n + 255) / 256;

    init_ws<<<(n * NCLS + 255) / 256, 256, 0, stream>>>(m1, dn1, agg1, m2, dn2, agg2, n);

    // ---- layer 1
    gemm1_wmma<<<gblocks, 128, 0, stream>>>(x, W1, h1, n);
    alpha_kernel<HID><<<nblocks, 256, 0, stream>>>(h1, a1s, a1d, as1, ad1, n);
    edge_max<<<eblocks, 256, 0, stream>>>(src, dst, as1, ad1, m1, E);
    edge_acc<HID><<<eblocks, 256, 0, stream>>>(src, dst, as1, ad1, m1, h1, dn1, agg1, E);
    finish1<<<nblocks, 256, 0, stream>>>(agg1, dn1, b1, n);

    // ---- layer 2
    gemm2_wmma<<<gblocks, 128, 0, stream>>>(agg1, W2, h2, n);
    alpha_kernel<NCLS><<<nblocks, 256, 0, stream>>>(h2, a2s, a2d, as2, ad2, n);
    edge_max<<<eblocks, 256, 0, stream>>>(src, dst, as2, ad2, m2, E);
    edge_acc<NCLS><<<eblocks, 256, 0, stream>>>(src, dst, as2, ad2, m2, h2, dn2, agg2, E);
    finish2<<<nblocks, 256, 0, stream>>>(agg2, dn2, b2, (float*)d_out, n);
}